// Roberta_modified_for_charEmb_438086664371
// MI455X (gfx1250) — compile-verified
//
#include <hip/hip_runtime.h>
#include <stdint.h>

// Problem constants (match reference): B=16, S=2048, D=768, fp32.
#define BB 16
#define SS 2048
#define DD 768

#define WAVES_PER_BLOCK  8
#define WORDS_PER_WAVE   4
#define WORDS_PER_BLOCK  (WAVES_PER_BLOCK * WORDS_PER_WAVE)   // 32
#define BLOCKS_PER_BATCH (SS / WORDS_PER_BLOCK)               // 64
#define NV4              (DD / (32 * 4))                      // 6 float4 per lane

typedef float v4f __attribute__((ext_vector_type(4)));

__global__ __launch_bounds__(256) void segmean_kernel(
    const float* __restrict__ x,      // [B, S, D]
    const int*   __restrict__ seg,    // [B, S] sorted per row
    float*       __restrict__ out)    // [B, S, D]
{
    __shared__ __align__(16) int seg_lds[SS];

    const int tid  = threadIdx.x;
    const int lane = tid & 31;
    const int wave = tid >> 5;

    const int blk       = blockIdx.x;
    const int b         = blk / BLOCKS_PER_BATCH;
    const int word_base = (blk % BLOCKS_PER_BATCH) * WORDS_PER_BLOCK;

    const int* seg_row = seg + (size_t)b * SS;

    // ---- Stage seg_ids[b, 0:2048] (8 KB) into LDS via CDNA5 async copy ----
    // 512 x b128; 256 threads issue 2 async loads each, then drain ASYNCcnt.
    {
        const uint32_t lds_base = (uint32_t)(uintptr_t)(&seg_lds[0]); // AS3 offset
        const uint32_t off0 = (uint32_t)tid * 16u;
        const uint32_t off1 = off0 + 4096u;
        const uint32_t l0 = lds_base + off0;
        const uint32_t l1 = lds_base + off1;
        asm volatile(
            "global_load_async_to_lds_b128 %0, %2, %4\n\t"
            "global_load_async_to_lds_b128 %1, %3, %4\n\t"
            "s_wait_asynccnt 0"
            :
            : "v"(l0), "v"(l1), "v"(off0), "v"(off1), "s"(seg_row)
            : "memory");
    }
    __syncthreads();

    // ---- Each wave handles WORDS_PER_WAVE output words (uniform per wave) ----
    for (int k = 0; k < WORDS_PER_WAVE; ++k) {
        const int w = word_base + wave * WORDS_PER_WAVE + k;

        // lower_bound(w) over sorted seg_lds[0:S)
        int first = 0, cnt = SS;
        while (cnt > 0) {
            int step = cnt >> 1;
            int idx  = first + step;
            if (seg_lds[idx] < w) { first = idx + 1; cnt -= step + 1; }
            else                  { cnt = step; }
        }
        const int lo = first;

        // lower_bound(w+1), restricted to [lo, S)
        cnt = SS - lo;
        while (cnt > 0) {
            int step = cnt >> 1;
            int idx  = first + step;
            if (seg_lds[idx] <= w) { first = idx + 1; cnt -= step + 1; }
            else                   { cnt = step; }
        }
        const int hi    = first;
        const int count = hi - lo;

        v4f acc[NV4];
#pragma unroll
        for (int i = 0; i < NV4; ++i) acc[i] = (v4f)0.0f;

        const float* xrow = x + ((size_t)b * SS + (size_t)lo) * DD;
        for (int r = 0; r < count; ++r) {
            const v4f* p = (const v4f*)xrow + lane;  // lane*16B within row
#pragma unroll
            for (int i = 0; i < NV4; ++i)
                acc[i] += __builtin_nontemporal_load(p + 32 * i);  // NT: row read once
            xrow += DD;
        }

        // mean; empty segment -> exact zeros (avoid 0*inf NaN)
        const float sc = (count > 0) ? (1.0f / (float)count) : 0.0f;
        v4f* o = (v4f*)(out + ((size_t)b * SS + (size_t)w) * DD) + lane;
#pragma unroll
        for (int i = 0; i < NV4; ++i)
            __builtin_nontemporal_store(acc[i] * sc, o + 32 * i);  // NT: write-once
    }
}

extern "C" void kernel_launch(void* const* d_in, const int* in_sizes, int n_in,
                              void* d_out, int out_size, void* d_ws, size_t ws_size,
                              hipStream_t stream) {
    (void)in_sizes; (void)n_in; (void)d_ws; (void)ws_size; (void)out_size;
    const float* x   = (const float*)d_in[0];   // robert_embed [16,2048,768] fp32
    const int*   seg = (const int*)d_in[1];     // seg_ids      [16,2048]     int32
    float*       out = (float*)d_out;           // [16,2048,768] fp32

    dim3 grid(BB * BLOCKS_PER_BATCH);  // 1024 workgroups
    dim3 block(WAVES_PER_BLOCK * 32);  // 256 threads = 8 wave32
    segmean_kernel<<<grid, block, 0, stream>>>(x, seg, out);
}